// ICGNNLayer_27865747816744
// MI455X (gfx1250) — compile-verified
//
#include <hip/hip_runtime.h>

typedef __attribute__((ext_vector_type(2))) float v2f;
typedef __attribute__((ext_vector_type(8))) float v8f;

#define D 64  // D_IN == D_OUT == D_U == 64

__device__ __forceinline__ float softplusf(float x) {
  // numerically stable softplus: max(x,0) + log1p(exp(-|x|))  (matches jax.nn.softplus)
  return fmaxf(x, 0.0f) + log1pf(expf(-fabsf(x)));
}

// ---------------------------------------------------------------------------
// Kernel 1: Wn = softplus(W), An = softplus(A)   (2 x 64x64, trivial)
// ---------------------------------------------------------------------------
__global__ void icgnn_softplus_kernel(const float* __restrict__ W,
                                      const float* __restrict__ A,
                                      float* __restrict__ Wn,
                                      float* __restrict__ An) {
  int i = blockIdx.x * blockDim.x + threadIdx.x;
  if (i < D * D) {
    Wn[i] = softplusf(W[i]);
    An[i] = softplusf(A[i]);
  }
}

// ---------------------------------------------------------------------------
// Kernel 2: Out[M,64] = X[M,64] @ Wn[64,64] (+ bias), full fp32 via
// V_WMMA_F32_16X16X4_F32. Block = 128 threads = 4 waves; block owns one
// 16-row tile, each wave owns one 16-col tile; K=64 -> 16 unrolled WMMAs.
//
// 32-bit A-matrix 16x4 layout (ISA 7.12.2): lanes 0-15 hold row M=lane,
// K={0,1} in VGPR{0,1}; lanes 16-31 hold K={2,3}.
// 32-bit C/D 16x16 layout: VGPR r holds M = r + 8*(lane>=16), N = lane&15.
// ---------------------------------------------------------------------------
__global__ void __launch_bounds__(128)
icgnn_gemm64_wmma_kernel(const float* __restrict__ X,
                         const float* __restrict__ Wn,
                         const float* __restrict__ bias,
                         float* __restrict__ Out,
                         int M) {
  __shared__ float sW[D * D];
  for (int i = threadIdx.x; i < D * D; i += blockDim.x) sW[i] = Wn[i];
  __syncthreads();

  const int wave = threadIdx.x >> 5;   // 0..3 -> column tile
  const int lane = threadIdx.x & 31;
  const int ly   = lane & 15;
  const int kh   = (lane >> 4) << 1;   // 0 for lanes 0-15, 2 for lanes 16-31
  const int n0   = wave << 4;

  const long long m0 = (long long)blockIdx.x << 4;
  long long row = m0 + ly;
  if (row > (long long)M - 1) row = (long long)M - 1;  // clamp: EXEC stays all-1s for WMMA
  const float* xrow = X + row * D;

  v8f acc = {};
#pragma unroll
  for (int k = 0; k < D; k += 4) {
    v2f a, b;
    a.x = xrow[k + kh];
    a.y = xrow[k + kh + 1];
    b.x = sW[(k + kh) * D + n0 + ly];
    b.y = sW[(k + kh + 1) * D + n0 + ly];
    // (neg_a, A, neg_b, B, c_mod, C, reuse_a, reuse_b)
    acc = __builtin_amdgcn_wmma_f32_16x16x4_f32(false, a, false, b,
                                                (short)0, acc, false, false);
  }

  const float badd  = bias ? bias[n0 + ly] : 0.0f;
  const int   mhalf = (lane >> 4) << 3;  // 0 or 8
#pragma unroll
  for (int r = 0; r < 8; ++r) {
    long long m = m0 + mhalf + r;
    if (m < M) Out[m * D + n0 + ly] = acc[r] + badd;
  }
}

// ---------------------------------------------------------------------------
// Kernel 3: edge scatter. One wave per edge: lane l loads float2 of
// H[src]*w (wave reads one contiguous 256B row -> coalesced b64 loads),
// then two native f32 atomic adds into Out[dst]. Relaxed/agent scope ->
// non-returning global_atomic_add_f32, no CAS loop.
// ---------------------------------------------------------------------------
__global__ void icgnn_edge_scatter_kernel(const float* __restrict__ H,
                                          const int* __restrict__ src,
                                          const int* __restrict__ dst,
                                          const float* __restrict__ ew,
                                          float* __restrict__ Out,
                                          int E) {
  const int gid  = blockIdx.x * blockDim.x + threadIdx.x;
  const int e    = gid >> 5;
  const int lane = threadIdx.x & 31;
  if (e >= E) return;

  const int   s = src[e];
  const int   d = dst[e];
  const float w = ew[e];

  const float2 hv = ((const float2*)(H + (long long)s * D))[lane];
  float* op = Out + (long long)d * D + (lane << 1);
  __hip_atomic_fetch_add(op,     hv.x * w, __ATOMIC_RELAXED, __HIP_MEMORY_SCOPE_AGENT);
  __hip_atomic_fetch_add(op + 1, hv.y * w, __ATOMIC_RELAXED, __HIP_MEMORY_SCOPE_AGENT);
}

// ---------------------------------------------------------------------------
// Kernel 4: out = relu(out), float4-vectorized (N*D divisible by 4)
// ---------------------------------------------------------------------------
__global__ void icgnn_relu_kernel(float* __restrict__ Out, int n4) {
  int i = blockIdx.x * blockDim.x + threadIdx.x;
  if (i < n4) {
    float4* p = (float4*)Out;
    float4 v = p[i];
    v.x = fmaxf(v.x, 0.0f);
    v.y = fmaxf(v.y, 0.0f);
    v.z = fmaxf(v.z, 0.0f);
    v.w = fmaxf(v.w, 0.0f);
    p[i] = v;
  }
}

// ---------------------------------------------------------------------------
// Launch. Inputs (setup_inputs order):
//   0: x [N,64] f32   1: edge_index [2,E] int   2: edge_weight [E] f32
//   3: u [N,64] f32   4: W [64,64] f32          5: A [64,64] f32
//   6: bias [64] f32
// ---------------------------------------------------------------------------
extern "C" void kernel_launch(void* const* d_in, const int* in_sizes, int n_in,
                              void* d_out, int out_size, void* d_ws, size_t ws_size,
                              hipStream_t stream) {
  (void)n_in; (void)out_size; (void)ws_size;

  const float* x    = (const float*)d_in[0];
  const int*   ei   = (const int*)d_in[1];
  const float* ew   = (const float*)d_in[2];
  const float* u    = (const float*)d_in[3];
  const float* W    = (const float*)d_in[4];
  const float* A    = (const float*)d_in[5];
  const float* bias = (const float*)d_in[6];
  float*       out  = (float*)d_out;

  const int N = in_sizes[0] / D;   // 50000
  const int E = in_sizes[2];       // 800000
  const int* src = ei;             // edge_index[0]
  const int* dst = ei + E;         // edge_index[1]

  // workspace layout: Wn (4096) | An (4096) | H (N*D)
  float* Wn = (float*)d_ws;
  float* An = Wn + D * D;
  float* H  = An + D * D;

  // 1) softplus weights
  icgnn_softplus_kernel<<<(D * D + 255) / 256, 256, 0, stream>>>(W, A, Wn, An);

  // 2) H = x @ Wn   (no bias)
  const int rowTiles = (N + 15) / 16;
  icgnn_gemm64_wmma_kernel<<<rowTiles, 128, 0, stream>>>(x, Wn, nullptr, H, N);

  // 3) out = u @ An + bias   (base for segment-sum)
  icgnn_gemm64_wmma_kernel<<<rowTiles, 128, 0, stream>>>(u, An, bias, out, N);

  // 4) out += scatter_add(H[src] * w -> dst)
  const long long sthreads = (long long)E * 32;
  const int sblocks = (int)((sthreads + 255) / 256);
  icgnn_edge_scatter_kernel<<<sblocks, 256, 0, stream>>>(H, src, dst, ew, out, E);

  // 5) out = relu(out)
  const int n4 = (N * D) / 4;
  icgnn_relu_kernel<<<(n4 + 255) / 256, 256, 0, stream>>>(out, n4);
}